// GraphNorm_1116691497446
// MI455X (gfx1250) — compile-verified
//
#include <hip/hip_runtime.h>
#include <stdint.h>

#ifndef __has_builtin
#define __has_builtin(x) 0
#endif

// ---------------------------------------------------------------------------
// GraphNorm for MI455X (gfx1250).
//  - batch_index is sorted -> segments are contiguous row ranges.
//  - seg_bounds_kernel: binary-search segment start offsets into d_ws.
//  - graphnorm_main: one workgroup (256 thr = 8 wave32) per segment.
//      Pass 1: stream segment rows HBM -> LDS with GLOBAL_LOAD_ASYNC_TO_LDS_B128
//              (double buffered, s_wait_asynccnt pipelining), accumulate
//              sum(x) and sum(x^2) per channel in one pass:
//              var = E[x^2] - 2*mm*E[x] + mm^2   (mm = mean*mean_scale)
//      Pass 2: re-read rows (L2-resident; NT loads = last use),
//              out = A*x + D with A=w/std, D=bias-A*mm, NT B128 stores.
//  Effective HBM traffic ~= 512MB read + 512MB write -> ~44us @ 23.3 TB/s.
// ---------------------------------------------------------------------------

typedef float f4 __attribute__((ext_vector_type(4)));
typedef int   v4i __attribute__((vector_size(4 * sizeof(int))));

#define CH        128    // channels
#define TROWS     16     // rows per staged tile (2 rows per wave)
#define NTHREADS  256    // 8 wave32
#define NWAVES    8
#define NSEG      1024   // B from the reference

// ---- gfx1250 async global->LDS copy (128-bit per lane) --------------------
__device__ __forceinline__ void async_copy_b128(const float* gsrc, float* ldst) {
#if __has_builtin(__builtin_amdgcn_global_load_async_to_lds_b128)
  __builtin_amdgcn_global_load_async_to_lds_b128(
      (__attribute__((address_space(1))) v4i*)(uintptr_t)gsrc,
      (__attribute__((address_space(3))) v4i*)(uint32_t)(uintptr_t)ldst,
      0, 0);
#else
  asm volatile("global_load_async_to_lds_b128 %0, %1, off"
               :: "v"((uint32_t)(uintptr_t)ldst),
                  "v"((uint64_t)(uintptr_t)gsrc)
               : "memory");
#endif
}

__device__ __forceinline__ void wait_async_le2() {
  asm volatile("s_wait_asynccnt 0x2" ::: "memory");
}
__device__ __forceinline__ void wait_async_0() {
  asm volatile("s_wait_asynccnt 0x0" ::: "memory");
}
__device__ __forceinline__ void wait_ds_0() {
  asm volatile("s_wait_dscnt 0x0" ::: "memory");
}

// ---- Main kernel: per-segment normalize (defined first for disasm) --------
__global__ __launch_bounds__(NTHREADS)
void graphnorm_main(const float* __restrict__ x,
                    const float* __restrict__ weight,
                    const float* __restrict__ bias,
                    const float* __restrict__ mscale,
                    const int* __restrict__ seg_start,
                    const int* __restrict__ nseg_p,
                    float* __restrict__ out) {
  __shared__ __align__(16) float tile[2][TROWS * CH];   // 16 KB double buffer
  __shared__ __align__(16) float red_s[NWAVES * CH];    // 4 KB
  __shared__ __align__(16) float red_q[NWAVES * CH];    // 4 KB
  __shared__ __align__(16) float coefA[CH];
  __shared__ __align__(16) float coefD[CH];

  const int b = blockIdx.x;
  if (b >= *nseg_p) return;
  const int start = seg_start[b];
  const int end   = seg_start[b + 1];
  const int cnt   = end - start;
  if (cnt <= 0) return;

  const int t    = threadIdx.x;
  const int lane = t & 31;
  const int g    = t >> 5;      // wave id 0..7
  const int col  = lane * 4;    // this lane's 4 channels

  const int T = (cnt + TROWS - 1) / TROWS;

  // Prologue: issue tile 0. Each wave stages its own rows {g, g+8} of the
  // tile (row index clamped to end-1 so EXEC stays full and ASYNCcnt is
  // deterministic: exactly 2 async-load instructions per wave per tile).
  {
    const int r0 = min(start + g,     end - 1);
    const int r1 = min(start + g + 8, end - 1);
    async_copy_b128(x + (size_t)r0 * CH + col, &tile[0][g * CH + col]);
    async_copy_b128(x + (size_t)r1 * CH + col, &tile[0][(g + 8) * CH + col]);
  }

  f4 acc_s = {0.f, 0.f, 0.f, 0.f};
  f4 acc_q = {0.f, 0.f, 0.f, 0.f};

  for (int k = 0; k < T; ++k) {
    const int row0 = start + k * TROWS;
    const int bk   = k & 1;
    if (k + 1 < T) {
      // Retire our previous LDS reads of the target buffer, issue the next
      // tile, then wait for *this* tile (<=2 of the 4 outstanding remain).
      wait_ds_0();
      const int nrow0 = row0 + TROWS;
      const int r0 = min(nrow0 + g,     end - 1);
      const int r1 = min(nrow0 + g + 8, end - 1);
      async_copy_b128(x + (size_t)r0 * CH + col, &tile[bk ^ 1][g * CH + col]);
      async_copy_b128(x + (size_t)r1 * CH + col, &tile[bk ^ 1][(g + 8) * CH + col]);
      wait_async_le2();
    } else {
      wait_async_0();
    }
    // Each wave consumes exactly the rows it staged: no cross-wave barrier.
    if (row0 + g < end) {
      const f4 v = *(const f4*)&tile[bk][g * CH + col];
      acc_s += v; acc_q += v * v;
    }
    if (row0 + g + 8 < end) {
      const f4 v = *(const f4*)&tile[bk][(g + 8) * CH + col];
      acc_s += v; acc_q += v * v;
    }
  }

  // Cross-wave reduction of per-channel sums.
  *(f4*)&red_s[g * CH + col] = acc_s;
  *(f4*)&red_q[g * CH + col] = acc_q;
  __syncthreads();

  if (t < CH) {
    float s = 0.f, q = 0.f;
#pragma unroll
    for (int w = 0; w < NWAVES; ++w) {
      s += red_s[w * CH + t];
      q += red_q[w * CH + t];
    }
    const float inv_cnt = 1.0f / (float)cnt;
    const float mean = s * inv_cnt;
    const float mm   = mean * mscale[t];                         // mean*mean_scale
    const float var  = q * inv_cnt - 2.0f * mm * mean + mm * mm; // E[(x-mm)^2]
    const float inv_std = 1.0f / sqrtf(var + 1e-6f);
    const float A = weight[t] * inv_std;
    coefA[t] = A;
    coefD[t] = bias[t] - A * mm;
  }
  __syncthreads();

  // Pass 2: out = A*x + D. Rows are L2-resident from pass 1. NT loads (last
  // use of x) and NT stores keep L2 free for other segments' reuse windows.
  const f4 a4 = *(const f4*)&coefA[col];
  const f4 d4 = *(const f4*)&coefD[col];
  for (int r = start + g; r < end; r += NWAVES) {
    const f4 v = __builtin_nontemporal_load((const f4*)(x + (size_t)r * CH + col));
    const f4 o = a4 * v + d4;
    __builtin_nontemporal_store(o, (f4*)(out + (size_t)r * CH + col));
  }
}

// ---- Segment bounds via lower_bound on sorted batch_index -----------------
__global__ void seg_bounds_kernel(const int* __restrict__ bidx, int n,
                                  const int* __restrict__ nseg_p,
                                  int* __restrict__ seg_start) {
  const int nseg = *nseg_p;
  const int b = blockIdx.x * blockDim.x + threadIdx.x;
  if (b > nseg) return;
  if (b == nseg) { seg_start[b] = n; return; }
  int lo = 0, hi = n;
  while (lo < hi) {
    const int mid = (lo + hi) >> 1;
    if (bidx[mid] < b) lo = mid + 1; else hi = mid;
  }
  seg_start[b] = lo;
}

// ---------------------------------------------------------------------------
extern "C" void kernel_launch(void* const* d_in, const int* in_sizes, int n_in,
                              void* d_out, int out_size, void* d_ws, size_t ws_size,
                              hipStream_t stream) {
  (void)n_in; (void)out_size; (void)ws_size;
  const float* x      = (const float*)d_in[0];
  const float* weight = (const float*)d_in[1];
  const float* bias   = (const float*)d_in[2];
  const float* mscale = (const float*)d_in[3];
  const int*   bidx   = (const int*)d_in[4];
  const int*   nsegp  = (const int*)d_in[5];
  const int    n      = in_sizes[4];

  int* seg_start = (int*)d_ws;  // (NSEG+1) ints

  seg_bounds_kernel<<<(NSEG + 1 + 255) / 256, 256, 0, stream>>>(bidx, n, nsegp, seg_start);
  graphnorm_main<<<NSEG, NTHREADS, 0, stream>>>(x, weight, bias, mscale,
                                                seg_start, nsegp, (float*)d_out);
}